// ChamferLoss_3212635537690
// MI455X (gfx1250) — compile-verified
//
#include <hip/hip_runtime.h>
#include <hip/hip_bf16.h>

typedef __attribute__((ext_vector_type(2))) float v2f;
typedef __attribute__((ext_vector_type(8))) float v8f;

#define CB 16      // batches
#define CN 4096    // points per cloud
#define TI 2       // 16-row A-tiles per wave (32 rows of X per wave)

// One wave owns TI*16 rows of X for one batch and streams all of Y in
// 16-column tiles, computing min_j ( rx_i + ry_j - 2<x_i,y_j> ) per row.
//
// Trick: V_WMMA_F32_16X16X4_F32 with the spare K=3 slot carrying the ry row:
//   A[m][0..2] = -2*x[m],  A[m][3] = 1.0
//   B[0..2][n] =  y[n],    B[3][n] = ry[n] = ||y[n]||^2
//   C = 0  =>  D[m][n] = ry[n] - 2<x[m],y[n]>        (rx added in epilogue)
//
// Layouts (f32, wave32):
//   A 16x4: lane m (0-15) holds A[m][0],A[m][1]; lane m+16 holds A[m][2],A[m][3]
//   B 4x16: lane n (0-15) holds B[0][n],B[1][n]; lane n+16 holds B[2][n],B[3][n]
//   C/D 16x16: vgpr r, lanes 0-15 -> (M=r, N=lane); lanes 16-31 -> (M=r+8)
__global__ __launch_bounds__(256)
void chamfer_rowmin_kernel(const float* __restrict__ X,
                           const float* __restrict__ Y,
                           float* __restrict__ rowmin,   // null -> atomic mode
                           float* __restrict__ out_atomic)
{
    const int lane = threadIdx.x & 31;
    const int wave = (blockIdx.x * (blockDim.x >> 5)) + (threadIdx.x >> 5);
    const int tilesPerBatch = CN / (16 * TI);            // 128
    const int b    = wave / tilesPerBatch;
    const int i0   = (wave % tilesPerBatch) * (16 * TI);

    const int  m   = lane & 15;
    const bool hi  = (lane >> 4) != 0;                   // holds K=2,3 half

    // ---- A tiles: each lane loads full point, computes rx locally ----
    v2f   a[TI];
    float rx[TI];
#pragma unroll
    for (int t = 0; t < TI; ++t) {
        const float* xp = X + ((size_t)b * CN + (size_t)(i0 + t * 16 + m)) * 3;
        float x0 = xp[0], x1 = xp[1], x2 = xp[2];
        rx[t]  = x0 * x0 + x1 * x1 + x2 * x2;            // rx[m] at every lane
        a[t].x = hi ? (-2.0f * x2) : (-2.0f * x0);
        a[t].y = hi ? 1.0f         : (-2.0f * x1);
    }

    float runmin[TI][8];
#pragma unroll
    for (int t = 0; t < TI; ++t)
#pragma unroll
        for (int r = 0; r < 8; ++r) runmin[t][r] = __builtin_inff();

    v8f czero;
#pragma unroll
    for (int r = 0; r < 8; ++r) czero[r] = 0.0f;         // loop-invariant C

    // ---- stream Y: no cross-lane ops, no C broadcast in the hot loop ----
    const float* yp = Y + (size_t)b * CN * 3 + (size_t)m * 3;
    for (int j0 = 0; j0 < CN; j0 += 16, yp += 48) {
        float y0 = yp[0], y1 = yp[1], y2 = yp[2];
        float ry = y0 * y0 + y1 * y1 + y2 * y2;

        v2f bm;
        bm.x = hi ? y2 : y0;
        bm.y = hi ? ry : y1;

#pragma unroll
        for (int t = 0; t < TI; ++t) {
            v8f d = __builtin_amdgcn_wmma_f32_16x16x4_f32(
                        false, a[t], false, bm, (short)0, czero, false, false);
#pragma unroll
            for (int r = 0; r < 8; ++r) runmin[t][r] = fminf(runmin[t][r], d[r]);
        }
    }

    // ---- epilogue: add rx[row], reduce min across the 16 columns ----
#pragma unroll
    for (int t = 0; t < TI; ++t) {
#pragma unroll
        for (int r = 0; r < 8; ++r) {
            // row handled by vgpr r in this half-wave: r + 8*hi; lane L holds rx[L&15]
            float rxRow = __shfl(rx[t], (hi ? 8 : 0) + r, 32);
            runmin[t][r] += rxRow;
        }
#pragma unroll
        for (int r = 0; r < 8; ++r) {
            float v = runmin[t][r];
            v = fminf(v, __shfl_xor(v, 1, 32));
            v = fminf(v, __shfl_xor(v, 2, 32));
            v = fminf(v, __shfl_xor(v, 4, 32));
            v = fminf(v, __shfl_xor(v, 8, 32));
            runmin[t][r] = v;  // lanes 0-15: rows t*16+0..7 ; lanes 16-31: +8
        }
    }

    if (rowmin != nullptr) {
        if (m == 0) {  // lanes 0 and 16 write 8 rows per tile
#pragma unroll
            for (int t = 0; t < TI; ++t) {
                float* dst = rowmin + (size_t)b * CN + i0 + t * 16 + (hi ? 8 : 0);
#pragma unroll
                for (int r = 0; r < 8; ++r) dst[r] = runmin[t][r];
            }
        }
    } else {
        float s = 0.0f;
#pragma unroll
        for (int t = 0; t < TI; ++t)
#pragma unroll
            for (int r = 0; r < 8; ++r) s += runmin[t][r];
        float stot = s + __shfl_xor(s, 16, 32);
        if (lane == 0) atomicAdd(out_atomic, stot);
    }
}

__global__ void reduce_sum_kernel(const float* __restrict__ vals,
                                  float* __restrict__ out, int n)
{
    __shared__ float sdata[256];
    float s = 0.0f;
    for (int i = threadIdx.x; i < n; i += 256) s += vals[i];
    sdata[threadIdx.x] = s;
    __syncthreads();
    for (int off = 128; off > 0; off >>= 1) {
        if ((int)threadIdx.x < off) sdata[threadIdx.x] += sdata[threadIdx.x + off];
        __syncthreads();
    }
    if (threadIdx.x == 0) out[0] = sdata[0];
}

__global__ void zero_kernel(float* __restrict__ out) { out[0] = 0.0f; }

extern "C" void kernel_launch(void* const* d_in, const int* in_sizes, int n_in,
                              void* d_out, int out_size, void* d_ws, size_t ws_size,
                              hipStream_t stream) {
    (void)in_sizes; (void)n_in; (void)out_size;
    const float* preds = (const float*)d_in[0];
    const float* gts   = (const float*)d_in[1];
    float* out = (float*)d_out;

    const int waves = CB * CN / (16 * TI);               // 2048 waves
    dim3 block(256);                                      // 8 waves / block
    dim3 grid(waves / 8);                                 // 256 blocks

    const size_t need = (size_t)2 * CB * CN * sizeof(float);  // 512 KB
    if (d_ws != nullptr && ws_size >= need) {
        float* mins = (float*)d_ws;
        // loss_2: rows = gts, min over preds columns
        chamfer_rowmin_kernel<<<grid, block, 0, stream>>>(gts, preds, mins, nullptr);
        // loss_1: rows = preds, min over gts columns (P transposed)
        chamfer_rowmin_kernel<<<grid, block, 0, stream>>>(preds, gts, mins + (size_t)CB * CN, nullptr);
        reduce_sum_kernel<<<1, 256, 0, stream>>>(mins, out, 2 * CB * CN);
    } else {
        zero_kernel<<<1, 1, 0, stream>>>(out);
        chamfer_rowmin_kernel<<<grid, block, 0, stream>>>(gts, preds, nullptr, out);
        chamfer_rowmin_kernel<<<grid, block, 0, stream>>>(preds, gts, nullptr, out);
    }
}